// UM_loss_52716428591263
// MI455X (gfx1250) — compile-verified
//
#include <hip/hip_runtime.h>
#include <math.h>

typedef __attribute__((ext_vector_type(2))) float v2f;
typedef __attribute__((ext_vector_type(4))) float v4f;
typedef __attribute__((ext_vector_type(8))) float v8f;

#define B_ 16
#define T_ 2048
#define D_ 2048
#define C_ 100
#define NSEG 4

// workspace layout (float offsets)
#define OFF_SUM_ACT 0                      // [NSEG][B_*D_] partial column sums
#define OFF_SUM_BKG (NSEG * B_ * D_)       // [NSEG][B_*D_]
#define OFF_NORMS   (2 * NSEG * B_ * D_)   // [B_*C_] sum_t (sup - pmask)^2
#define OFF_PCNT    (OFF_NORMS + B_ * C_)  // [B_*C_] sum_t pmask
#define OFF_UM      (OFF_PCNT + B_ * C_)   // [B_]
#define OFF_ST      (OFF_UM + B_)          // [1024] MSE partials
#define OFF_SC      (OFF_ST + 1024)        // scalars: [0]=cls sum, [1]=be sum
#define N_ST_BLOCKS 1024

__device__ __forceinline__ float block_reduce_256(float v, float* smem) {
  int tid = threadIdx.x;
  smem[tid] = v;
  __syncthreads();
  #pragma unroll
  for (int s = 128; s > 0; s >>= 1) {
    if (tid < s) smem[tid] += smem[tid + s];
    __syncthreads();
  }
  float r = smem[0];
  __syncthreads();
  return r;
}

// ---------------------------------------------------------------------------
// Kernel A: time-axis column sums via WMMA f32 16x16x4 with A = ones.
// Since every row of D equals the column sums, K-slot placement is free:
// lane l feeds column l%16 of any WMMA it participates in. So each lane
// loads a float4 (4 adjacent columns) per row and scatters the components
// into 4 accumulators -> 64-wide d-tile per wave, b128 loads/stores.
// Wave = (tensor s, batch b, 64-wide d-tile dt, T-segment seg):
//   2 * 16 * 32 * 4 = 4096 waves = 512 blocks x 8 waves (exact cover,
//   EXEC all-ones for WMMA). Each wave reduces 512 rows in 4-row chunks:
//   half 0 (lanes 0-15) loads rows t,t+1; half 1 rows t+2,t+3.
// ---------------------------------------------------------------------------
__global__ void feat_colsum_wmma(const float* __restrict__ feat_act,
                                 const float* __restrict__ feat_bkg,
                                 float* __restrict__ ws) {
  const int wave = blockIdx.x * (blockDim.x >> 5) + (threadIdx.x >> 5);
  const int lane = threadIdx.x & 31;

  const int seg = wave & 3;
  const int dt  = (wave >> 2) & 31;
  const int b   = (wave >> 7) & 15;
  const int s   = wave >> 11;

  const int m    = lane & 15;
  const int half = lane >> 4;
  const int col  = dt * 64 + 4 * m;
  const int trow = seg * (T_ / NSEG) + half * 2;

  const float* f = s ? feat_bkg : feat_act;
  const float* p = f + ((size_t)b * T_ + trow) * D_ + col;

  v2f a; a.x = 1.0f; a.y = 1.0f;           // A = all ones
  v8f c0 = {}, c1 = {}, c2 = {}, c3 = {};

  #pragma unroll 2
  for (int it = 0; it < (T_ / NSEG); it += 4) {
    const float* q = p + (size_t)it * D_;
    v4f r0 = *(const v4f*)(q);             // row trow+it
    v4f r1 = *(const v4f*)(q + D_);        // row trow+it+1
    v2f b0; b0.x = r0.x; b0.y = r1.x;
    v2f b1; b1.x = r0.y; b1.y = r1.y;
    v2f b2; b2.x = r0.z; b2.y = r1.z;
    v2f b3; b3.x = r0.w; b3.y = r1.w;
    c0 = __builtin_amdgcn_wmma_f32_16x16x4_f32(false, a, false, b0, (short)0, c0, false, false);
    c1 = __builtin_amdgcn_wmma_f32_16x16x4_f32(false, a, false, b1, (short)0, c1, false, false);
    c2 = __builtin_amdgcn_wmma_f32_16x16x4_f32(false, a, false, b2, (short)0, c2, false, false);
    c3 = __builtin_amdgcn_wmma_f32_16x16x4_f32(false, a, false, b3, (short)0, c3, false, false);
  }

  // D layout: VGPR0 of lanes 0-15 holds D[0][lane] = column-sum for that lane
  if (lane < 16) {
    float* dst = ws + (s ? OFF_SUM_BKG : OFF_SUM_ACT) + seg * (B_ * D_);
    v4f o; o.x = c0[0]; o.y = c1[0]; o.z = c2[0]; o.w = c3[0];
    *(v4f*)(dst + (size_t)b * D_ + dt * 64 + 4 * lane) = o;
  }
}

// ---------------------------------------------------------------------------
// Kernel B: fold segment partials, per-batch L2 norms -> um term per batch.
// ---------------------------------------------------------------------------
__global__ void um_norms(float* __restrict__ ws) {
  __shared__ float smem[256];
  const int b = blockIdx.x;
  const float invT = 1.0f / (float)T_;
  float sa = 0.f, sb = 0.f;
  for (int d = threadIdx.x; d < D_; d += 256) {
    float va = 0.f, vb = 0.f;
    #pragma unroll
    for (int g = 0; g < NSEG; ++g) {
      va += ws[OFF_SUM_ACT + g * (B_ * D_) + b * D_ + d];
      vb += ws[OFF_SUM_BKG + g * (B_ * D_) + b * D_ + d];
    }
    va *= invT; vb *= invT;
    sa += va * va;
    sb += vb * vb;
  }
  sa = block_reduce_256(sa, smem);
  sb = block_reduce_256(sb, smem);
  if (threadIdx.x == 0) {
    float act_norm = sqrtf(sa);
    float bkg_norm = sqrtf(sb);
    float la = fmaxf(100.0f - act_norm, 0.0f);
    float t = la + bkg_norm;
    ws[OFF_UM + b] = t * t;
  }
}

// ---------------------------------------------------------------------------
// Kernel C: per-(b,c) sum_t (sup - pmask)^2 and sum_t pmask.
// Threads laid out as 2 t-phases x 128 c-lanes: loads are c-contiguous.
// ---------------------------------------------------------------------------
__global__ void sup_norms(const float* __restrict__ gt,
                          const float* __restrict__ sup,
                          float* __restrict__ ws) {
  __shared__ float sn[256];
  __shared__ float sp[256];
  const int b  = blockIdx.x;
  const int r  = threadIdx.x >> 7;   // 0 or 1
  const int cl = threadIdx.x & 127;
  float accn = 0.f, accp = 0.f;
  if (cl < C_) {
    const size_t base = (size_t)b * T_ * C_;
    for (int t = r; t < T_; t += 2) {
      size_t idx = base + (size_t)t * C_ + cl;
      float g  = gt[idx];
      float pm = g > 0.5f ? 1.0f : 0.0f;
      float d  = sup[idx] - pm;
      accn += d * d;
      accp += pm;
    }
  }
  sn[threadIdx.x] = accn;
  sp[threadIdx.x] = accp;
  __syncthreads();
  if (threadIdx.x < C_) {
    ws[OFF_NORMS + b * C_ + threadIdx.x] = sn[threadIdx.x] + sn[threadIdx.x + 128];
    ws[OFF_PCNT  + b * C_ + threadIdx.x] = sp[threadIdx.x] + sp[threadIdx.x + 128];
  }
}

// ---------------------------------------------------------------------------
// Kernel D: classification + background BCE sums (1600 elements, one block).
// ---------------------------------------------------------------------------
__global__ void cls_be(const float* __restrict__ score_act,
                       const float* __restrict__ score_bkg,
                       const float* __restrict__ label,
                       float* __restrict__ ws) {
  __shared__ float smem[256];
  __shared__ float rowsum[B_];
  if (threadIdx.x < B_) {
    float s = 0.f;
    for (int c = 0; c < C_; ++c) s += label[threadIdx.x * C_ + c];
    rowsum[threadIdx.x] = s;
  }
  __syncthreads();
  const float EPS = 1e-7f;
  const float tb  = 1.0f / (float)C_;
  float accc = 0.f, accb = 0.f;
  for (int i = threadIdx.x; i < B_ * C_; i += 256) {
    int b = i / C_;
    float t = label[i] / rowsum[b];
    float p = fminf(fmaxf(score_act[i], EPS), 1.0f - EPS);
    accc += t * logf(p) + (1.0f - t) * log1pf(-p);
    float q = fminf(fmaxf(score_bkg[i], EPS), 1.0f - EPS);
    accb += tb * logf(q) + (1.0f - tb) * log1pf(-q);
  }
  accc = block_reduce_256(accc, smem);
  accb = block_reduce_256(accb, smem);
  if (threadIdx.x == 0) { ws[OFF_SC + 0] = accc; ws[OFF_SC + 1] = accb; }
}

// ---------------------------------------------------------------------------
// Kernel E: student/teacher MSE partial sums, float4 streams.
// ---------------------------------------------------------------------------
__global__ void st_partial(const float* __restrict__ cs,
                           const float* __restrict__ ct,
                           float* __restrict__ ws) {
  __shared__ float smem[256];
  const int N4 = (B_ * T_ * C_) / 4;   // 819200, exact
  const v4f* cs4 = (const v4f*)cs;
  const v4f* ct4 = (const v4f*)ct;
  float acc = 0.f;
  for (int i = blockIdx.x * 256 + threadIdx.x; i < N4; i += N_ST_BLOCKS * 256) {
    v4f a = cs4[i];
    v4f b = ct4[i];
    float d0 = a.x - b.x, d1 = a.y - b.y, d2 = a.z - b.z, d3 = a.w - b.w;
    acc += d0 * d0 + d1 * d1 + d2 * d2 + d3 * d3;
  }
  acc = block_reduce_256(acc, smem);
  if (threadIdx.x == 0) ws[OFF_ST + blockIdx.x] = acc;
}

// ---------------------------------------------------------------------------
// Kernel F: finalize all six losses.
// ---------------------------------------------------------------------------
__global__ void finalize(float* __restrict__ ws, float* __restrict__ out) {
  __shared__ float smem[256];
  float st = 0.f;
  for (int i = threadIdx.x; i < N_ST_BLOCKS; i += 256) st += ws[OFF_ST + i];
  st = block_reduce_256(st, smem);

  float sup_sum = 0.f, vcnt = 0.f;
  for (int i = threadIdx.x; i < B_ * C_; i += 256) {
    if (ws[OFF_PCNT + i] > 0.f) {
      sup_sum += sqrtf(ws[OFF_NORMS + i]);
      vcnt    += 1.f;
    }
  }
  sup_sum = block_reduce_256(sup_sum, smem);
  vcnt    = block_reduce_256(vcnt, smem);

  float um = 0.f;
  if (threadIdx.x < B_) um = ws[OFF_UM + threadIdx.x];
  um = block_reduce_256(um, smem);

  if (threadIdx.x == 0) {
    float loss_st  = st / (float)(B_ * T_ * C_);
    float cnt      = fmaxf(vcnt, 1.0f);
    float loss_sup = sup_sum / cnt;
    float loss_um  = um / (float)B_;
    float loss_cls = -ws[OFF_SC + 0] / (float)(B_ * C_);
    float loss_be  = -ws[OFF_SC + 1] / (float)(B_ * C_);
    float total = loss_cls + 0.0005f * loss_um + 1.0f * loss_be
                + 1.0f * loss_sup + 1.0f * loss_st;
    out[0] = total;    out[1] = loss_cls; out[2] = loss_be;
    out[3] = loss_um;  out[4] = loss_sup; out[5] = loss_st;
  }
}

extern "C" void kernel_launch(void* const* d_in, const int* in_sizes, int n_in,
                              void* d_out, int out_size, void* d_ws, size_t ws_size,
                              hipStream_t stream) {
  const float* score_act = (const float*)d_in[0];
  const float* score_bkg = (const float*)d_in[1];
  const float* feat_act  = (const float*)d_in[2];
  const float* feat_bkg  = (const float*)d_in[3];
  const float* label     = (const float*)d_in[4];
  const float* gt        = (const float*)d_in[5];
  const float* sup_cas   = (const float*)d_in[6];
  const float* cas_s     = (const float*)d_in[7];
  const float* cas_t     = (const float*)d_in[8];
  float* ws  = (float*)d_ws;
  float* out = (float*)d_out;

  // 4096 waves: 2 tensors * 16 batch * 32 d-tiles(64-wide) * 4 T-segments
  feat_colsum_wmma<<<512, 256, 0, stream>>>(feat_act, feat_bkg, ws);
  um_norms<<<B_, 256, 0, stream>>>(ws);
  sup_norms<<<B_, 256, 0, stream>>>(gt, sup_cas, ws);
  cls_be<<<1, 256, 0, stream>>>(score_act, score_bkg, label, ws);
  st_partial<<<N_ST_BLOCKS, 256, 0, stream>>>(cas_s, cas_t, ws);
  finalize<<<1, 256, 0, stream>>>(ws, out);
}